// dgl_gin_1099511628221
// MI455X (gfx1250) — compile-verified
//
#include <hip/hip_runtime.h>

typedef __attribute__((ext_vector_type(2))) float v2f;
typedef __attribute__((ext_vector_type(8))) float v8f;

#define N_NODES 50000
#define N_EDGES 800000
#define IN_DIM  128
#define HIDDEN  128
#define CLASSES 64

// Y[N,H] = X[N,K] @ W[H,K]^T  (no bias), one 16x16 output tile per wave,
// K-loop in steps of 4 using v_wmma_f32_16x16x4_f32 (full fp32 precision).
//
// A-frag (16x4 f32): lane<16 holds K pair (k0,k0+1) of row M=lane&15,
//                    lane>=16 holds pair (k0+2,k0+3)       -> contiguous float2.
// B-frag (4x16 f32): same pattern with N=lane&15 from W rows -> contiguous float2.
// D (16x16 f32, 8 VGPRs): row = 8*(lane>>4)+vr, col = lane&15.
__global__ void __launch_bounds__(256) gemm_xwt_wmma(
    const float* __restrict__ X, const float* __restrict__ W,
    float* __restrict__ Y, int N, int K, int H, int totalTiles) {
  const int lane = threadIdx.x & 31;
  const int wave = threadIdx.x >> 5;
  const int t = blockIdx.x * 8 + wave;          // 8 waves / block
  if (t >= totalTiles) return;                  // wave-uniform; EXEC all-1 past here
  const int nTiles = H >> 4;
  const int mTile = t / nTiles;
  const int nTile = t - mTile * nTiles;
  const int r  = lane & 15;
  const int kh = (lane >> 4) << 1;              // 0 or 2
  const float* xrow = X + (size_t)(mTile * 16 + r) * K + kh;
  const float* wrow = W + (size_t)(nTile * 16 + r) * K + kh;
  v8f acc = {};
#pragma unroll 8
  for (int k0 = 0; k0 < K; k0 += 4) {
    v2f a = *(const v2f*)(xrow + k0);
    v2f b = *(const v2f*)(wrow + k0);
    acc = __builtin_amdgcn_wmma_f32_16x16x4_f32(
        /*neg_a=*/false, a, /*neg_b=*/false, b,
        /*c_mod=*/(short)0, acc, /*reuse_a=*/false, /*reuse_b=*/false);
  }
  float* yout = Y + (size_t)(mTile * 16 + (lane >> 4) * 8) * H + nTile * 16 + r;
#pragma unroll
  for (int vr = 0; vr < 8; ++vr) yout[(size_t)vr * H] = acc[vr];
}

__global__ void zero_f4(float4* __restrict__ p, int n4) {
  int i = blockIdx.x * blockDim.x + threadIdx.x;
  if (i < n4) p[i] = make_float4(0.f, 0.f, 0.f, 0.f);
}

// agg[dst[e]] += Y[src[e]]  (D floats per edge). Each thread handles 4
// contiguous floats; with shift=log2(D/4) a wave covers exactly one edge
// (D=128) -> scalar src/dst loads + coalesced 512B row gather; atomics land
// in L2 (node tables are L2-resident: <= 25.6 MB << 192 MB).
__global__ void edge_scatter(const float* __restrict__ Y,
                             const int* __restrict__ src,
                             const int* __restrict__ dst,
                             float* __restrict__ agg,
                             int nEdges, int D, int shift) {
  int tid = blockIdx.x * blockDim.x + threadIdx.x;
  int e = tid >> shift;
  if (e >= nEdges) return;
  int c = (tid & ((1 << shift) - 1)) << 2;
  int s = src[e], d = dst[e];
  const float4 v = *(const float4*)(Y + (size_t)s * D + c);
  float* p = agg + (size_t)d * D + c;
  atomicAdd(p + 0, v.x);
  atomicAdd(p + 1, v.y);
  atomicAdd(p + 2, v.z);
  atomicAdd(p + 3, v.w);
}

// out = elu(a0*(y + aggA + b)) + elu(a1*(y + aggB + b)), elu alpha=1.
__global__ void combine_elu(const float* __restrict__ Yself,
                            const float* __restrict__ aggA,
                            const float* __restrict__ aggB,
                            const float* __restrict__ bias,
                            const float* __restrict__ attn,
                            float* __restrict__ out, int total, int dMask) {
  int i = blockIdx.x * blockDim.x + threadIdx.x;
  if (i >= total) return;
  float a0 = attn[0], a1 = attn[1];
  float base = Yself[i] + bias[i & dMask];
  float z0 = a0 * (base + aggA[i]);
  float z1 = a1 * (base + aggB[i]);
  float e0 = z0 > 0.0f ? z0 : (expf(z0) - 1.0f);
  float e1 = z1 > 0.0f ? z1 : (expf(z1) - 1.0f);
  out[i] = e0 + e1;
}

extern "C" void kernel_launch(void* const* d_in, const int* in_sizes, int n_in,
                              void* d_out, int out_size, void* d_ws, size_t ws_size,
                              hipStream_t stream) {
  const float* features = (const float*)d_in[0];
  const float* W1   = (const float*)d_in[1];
  const float* b1   = (const float*)d_in[2];
  const float* W2   = (const float*)d_in[3];
  const float* b2   = (const float*)d_in[4];
  const float* attn = (const float*)d_in[5];
  const int* src1 = (const int*)d_in[6];
  const int* dst1 = (const int*)d_in[7];
  const int* src2 = (const int*)d_in[8];
  const int* dst2 = (const int*)d_in[9];
  float* out = (float*)d_out;
  float* ws  = (float*)d_ws;

  const int N = N_NODES, E = N_EDGES;

  // Workspace layout (3 * N*128 floats = 76.8 MB):
  float* buf0 = ws;                      // y1, then x1 (combine writes in place)
  float* buf1 = ws + (size_t)N * 128;    // aggA; layer 2: y2 (first N*64) + aggC
  float* buf2 = ws + (size_t)N * 256;    // aggB; layer 2: aggD (first N*64)
  float* y2   = buf1;
  float* aggC = buf1 + (size_t)N * 64;   // buf1 second half; contiguous with aggD
  float* aggD = buf2;

  // ---- layer 1 ----
  const int tiles1 = (N / 16) * (HIDDEN / 16);          // 25000 tiles
  gemm_xwt_wmma<<<(tiles1 + 7) / 8, 256, 0, stream>>>(
      features, W1, buf0, N, IN_DIM, HIDDEN, tiles1);

  {
    int n4 = N * 256 / 4;  // zero aggA + aggB (contiguous)
    zero_f4<<<(n4 + 255) / 256, 256, 0, stream>>>((float4*)buf1, n4);
  }
  {
    int threads = E * (128 / 4);
    edge_scatter<<<(threads + 255) / 256, 256, 0, stream>>>(buf0, src1, dst1, buf1, E, 128, 5);
    edge_scatter<<<(threads + 255) / 256, 256, 0, stream>>>(buf0, src2, dst2, buf2, E, 128, 5);
  }
  {
    int total = N * HIDDEN;
    combine_elu<<<(total + 255) / 256, 256, 0, stream>>>(
        buf0, buf1, buf2, b1, attn, buf0, total, HIDDEN - 1);
  }

  // ---- layer 2 ----
  const int tiles2 = (N / 16) * (CLASSES / 16);         // 12500 tiles
  gemm_xwt_wmma<<<(tiles2 + 7) / 8, 256, 0, stream>>>(
      buf0, W2, y2, N, HIDDEN, CLASSES, tiles2);

  {
    int n4 = N * 128 / 4;  // aggC (buf1 2nd half) and aggD (buf2 1st half) are contiguous
    zero_f4<<<(n4 + 255) / 256, 256, 0, stream>>>((float4*)aggC, n4);
  }
  {
    int threads = E * (64 / 4);
    edge_scatter<<<(threads + 255) / 256, 256, 0, stream>>>(y2, src1, dst1, aggC, E, 64, 4);
    edge_scatter<<<(threads + 255) / 256, 256, 0, stream>>>(y2, src2, dst2, aggD, E, 64, 4);
  }
  {
    int total = N * CLASSES;
    combine_elu<<<(total + 255) / 256, 256, 0, stream>>>(
        y2, aggC, aggD, b2, attn, out, total, CLASSES - 1);
  }
}